// WeightedMMD_36756330119753
// MI455X (gfx1250) — compile-verified
//
#include <hip/hip_runtime.h>
#include <hip/hip_bf16.h>

typedef __attribute__((ext_vector_type(2))) float v2f;
typedef __attribute__((ext_vector_type(8))) float v8f;
typedef __attribute__((ext_vector_type(4))) int   v4i;

#define GLOBAL_AS __attribute__((address_space(1)))
#define LDS_AS    __attribute__((address_space(3)))

#define DD    8192      // feature dim / W dim
#define NROW  100       // rows of source/target
#define NPAD  112       // 7 tiles of 16
#define GSZ   (NPAD*NPAD)

// workspace layout (floats)
#define OFF_PART 0            // 512 frob partials (128 blocks * 4 waves)
#define OFF_G    512          // 3 * 112*112 Gram buffers
#define OFF_DIFF 38144        // 112*8192 padded diff  (38144 % 16 == 0)

// async global->LDS staging if the gfx1250 builtins are available
#if __has_builtin(__builtin_amdgcn_global_load_async_to_lds_b128) && \
    __has_builtin(__builtin_amdgcn_s_wait_asynccnt)
#define ASYNC_LDS 1
#else
#define ASYNC_LDS 0
#endif

// ---------------------------------------------------------------------------
// Kernel 1: diffp[112][8192] = source - target, rows 100..111 zeroed
// ---------------------------------------------------------------------------
__global__ void wmmd_diff_kernel(const float* __restrict__ src,
                                 const float* __restrict__ tgt,
                                 float* __restrict__ diffp) {
    int idx = blockIdx.x * 256 + threadIdx.x;      // 0 .. 112*8192-1
    int row = idx >> 13;                           // /8192
    float v = 0.0f;
    if (row < NROW) v = src[idx] - tgt[idx];
    diffp[idx] = v;
}

// ---------------------------------------------------------------------------
// Kernel 2: Gram matrices G_ss, G_tt, G_st (112x112 each, pad region garbage,
//           masked later).  One 16x16 C tile per wave, K-loop of f32 WMMA.
//           grid = 3*49 blocks of 32 threads (one wave).
// ---------------------------------------------------------------------------
__global__ void wmmd_gram_kernel(const float* __restrict__ S,
                                 const float* __restrict__ T,
                                 float* __restrict__ G) {
    const int lane   = threadIdx.x;        // 0..31, full wave
    const int lane15 = lane & 15;
    const int khalf  = lane >> 4;

    const int mat = blockIdx.x / 49;       // 0: S·S^T  1: T·T^T  2: S·T^T
    const int rem = blockIdx.x % 49;
    const int ti  = rem / 7;
    const int tj  = rem % 7;

    const float* Am = (mat == 1) ? T : S;
    const float* Bm = (mat == 0) ? S : T;

    int rowA = ti * 16 + lane15; if (rowA >= NROW) rowA = NROW - 1;   // clamp pad
    int rowB = tj * 16 + lane15; if (rowB >= NROW) rowB = NROW - 1;

    const float* pa = Am + (size_t)rowA * DD + khalf * 2;
    const float* pb = Bm + (size_t)rowB * DD + khalf * 2;

    v8f acc0 = {};
    v8f acc1 = {};
#pragma unroll 4
    for (int k = 0; k < DD; k += 8) {
        v2f a0 = *(const v2f*)(pa + k);
        v2f b0 = *(const v2f*)(pb + k);
        acc0 = __builtin_amdgcn_wmma_f32_16x16x4_f32(false, a0, false, b0,
                                                     (short)0, acc0, false, false);
        v2f a1 = *(const v2f*)(pa + k + 4);
        v2f b1 = *(const v2f*)(pb + k + 4);
        acc1 = __builtin_amdgcn_wmma_f32_16x16x4_f32(false, a1, false, b1,
                                                     (short)0, acc1, false, false);
    }
    v8f acc = acc0 + acc1;

    float* gp = G + (size_t)mat * GSZ;
#pragma unroll
    for (int v = 0; v < 8; ++v) {
        int r = ti * 16 + khalf * 8 + v;       // C/D layout: lanes16-31 => M+8
        int c = tj * 16 + lane15;
        gp[(size_t)r * NPAD + c] = acc[v];
    }
}

// ---------------------------------------------------------------------------
// Kernel 3: frob GEMM  C = diffp(112x8192) @ W(8192x8192), sum of C^2.
//   grid = 128 blocks * 128 threads (4 waves). Block owns 64 columns of W,
//   wave owns 16. A staged in LDS per K-chunk of 64 (row stride 68 floats),
//   via async global->LDS copies when available.
//   Deterministic: one partial per wave -> ws[OFF_PART + blk*4 + wave].
// ---------------------------------------------------------------------------
#define KC   64
#define LDA  68      // padded LDS row stride (floats)

__global__ void wmmd_frob_kernel(const float* __restrict__ diffp,
                                 const float* __restrict__ W,
                                 float* __restrict__ part) {
    __shared__ float ldsA[NPAD * LDA];

    const int tid    = threadIdx.x;        // 0..127
    const int wave   = tid >> 5;           // 0..3
    const int lane   = tid & 31;
    const int lane15 = lane & 15;
    const int khalf  = lane >> 4;
    const int n0     = blockIdx.x * 64 + wave * 16;

    v8f acc[7] = {};

    for (int kb = 0; kb < DD; kb += KC) {
        __syncthreads();   // previous chunk's LDS reads done before overwrite
        // cooperative A load: 112*16 float4 / 128 threads = 14 each
#if ASYNC_LDS
#pragma unroll
        for (int t = 0; t < 14; ++t) {
            int idx = tid + t * 128;
            int row = idx >> 4;
            int c4  = idx & 15;
            const float* g = diffp + (size_t)row * DD + kb + c4 * 4;
            __builtin_amdgcn_global_load_async_to_lds_b128(
                (GLOBAL_AS v4i*)g,
                (LDS_AS v4i*)&ldsA[row * LDA + c4 * 4],
                0, 0);
        }
        __builtin_amdgcn_s_wait_asynccnt(0);
#else
#pragma unroll
        for (int t = 0; t < 14; ++t) {
            int idx = tid + t * 128;
            int row = idx >> 4;
            int c4  = idx & 15;
            float4 val = *(const float4*)(diffp + (size_t)row * DD + kb + c4 * 4);
            *(float4*)(&ldsA[row * LDA + c4 * 4]) = val;
        }
#endif
        __syncthreads();

#pragma unroll
        for (int kk = 0; kk < KC; kk += 4) {
            const int kcol = kk + khalf * 2;
            const float* wp = W + (size_t)(kb + kcol) * DD + n0 + lane15;
            v2f b;
            b.x = wp[0];
            b.y = wp[DD];
            if (kb + KC < DD)   // prefetch next K-chunk of W
                __builtin_prefetch(wp + (size_t)KC * DD, 0, 1);
#pragma unroll
            for (int mt = 0; mt < 7; ++mt) {
                v2f a = *(const v2f*)(&ldsA[(mt * 16 + lane15) * LDA + kcol]);
                acc[mt] = __builtin_amdgcn_wmma_f32_16x16x4_f32(
                    false, a, false, b, (short)0, acc[mt], false, false);
            }
        }
    }

    // pad rows of diffp are zero -> pad rows of C are zero; just square+sum all
    float s = 0.0f;
#pragma unroll
    for (int mt = 0; mt < 7; ++mt)
#pragma unroll
        for (int v = 0; v < 8; ++v)
            s += acc[mt][v] * acc[mt][v];

#pragma unroll
    for (int o = 16; o > 0; o >>= 1)
        s += __shfl_xor(s, o, 32);

    if (lane == 0) part[blockIdx.x * 4 + wave] = s;
}

// ---------------------------------------------------------------------------
// Kernel 4: final reduction. triu exp-sums over the 3 Gram buffers (i,j<100)
//           + sum of 512 frob partials -> scalar.
// ---------------------------------------------------------------------------
__global__ void wmmd_final_kernel(const float* __restrict__ G,
                                  const float* __restrict__ part,
                                  float* __restrict__ out) {
    const int tid = threadIdx.x;           // 256 threads, 1 block
    const float* Gss = G;
    const float* Gtt = G + GSZ;
    const float* Gst = G + 2 * GSZ;

    float sss = 0.0f, stt = 0.0f, sst = 0.0f, fro = 0.0f;

    for (int idx = tid; idx < NROW * NROW; idx += 256) {
        int i = idx / NROW;
        int j = idx % NROW;
        if (i <= j) {
            float ssi = Gss[i * NPAD + i];
            float ssj = Gss[j * NPAD + j];
            float tti = Gtt[i * NPAD + i];
            float ttj = Gtt[j * NPAD + j];
            sss += expf(-(ssi + ssj - 2.0f * Gss[i * NPAD + j]) * 0.5f);
            stt += expf(-(tti + ttj - 2.0f * Gtt[i * NPAD + j]) * 0.5f);
            sst += expf(-(ssi + ttj - 2.0f * Gst[i * NPAD + j]) * 0.5f);
        }
    }
    for (int idx = tid; idx < 512; idx += 256) fro += part[idx];

#pragma unroll
    for (int o = 16; o > 0; o >>= 1) {
        sss += __shfl_xor(sss, o, 32);
        stt += __shfl_xor(stt, o, 32);
        sst += __shfl_xor(sst, o, 32);
        fro += __shfl_xor(fro, o, 32);
    }

    __shared__ float sm[4][8];
    int wid = tid >> 5;
    if ((tid & 31) == 0) {
        sm[0][wid] = sss; sm[1][wid] = stt; sm[2][wid] = sst; sm[3][wid] = fro;
    }
    __syncthreads();
    if (tid == 0) {
        float a = 0, b = 0, c = 0, f = 0;
        for (int w = 0; w < 8; ++w) { a += sm[0][w]; b += sm[1][w]; c += sm[2][w]; f += sm[3][w]; }
        const float scale = 2.0f / ((float)NROW * (float)NROW);
        float mmd = scale * (a + b - 2.0f * c);
        out[0] = mmd + f;
    }
}

// ---------------------------------------------------------------------------
extern "C" void kernel_launch(void* const* d_in, const int* in_sizes, int n_in,
                              void* d_out, int out_size, void* d_ws, size_t ws_size,
                              hipStream_t stream) {
    const float* src = (const float*)d_in[0];
    const float* tgt = (const float*)d_in[1];
    const float* W   = (const float*)d_in[2];
    float* out = (float*)d_out;
    float* ws  = (float*)d_ws;

    float* part  = ws + OFF_PART;
    float* G     = ws + OFF_G;
    float* diffp = ws + OFF_DIFF;

    // 1) diff + zero-pad
    wmmd_diff_kernel<<<(NPAD * DD) / 256, 256, 0, stream>>>(src, tgt, diffp);
    // 2) Gram matrices (3 x 7x7 tiles, one wave each)
    wmmd_gram_kernel<<<3 * 49, 32, 0, stream>>>(src, tgt, G);
    // 3) frob GEMM with LDS-staged A
    wmmd_frob_kernel<<<DD / 64, 128, 0, stream>>>(diffp, W, part);
    // 4) finalize scalar
    wmmd_final_kernel<<<1, 256, 0, stream>>>(G, part, out);
}